// Colorcal_TwoDatasets_6536940224722
// MI455X (gfx1250) — compile-verified
//
#include <hip/hip_runtime.h>
#include <hip/hip_bf16.h>

// Problem constants (fixed by the reference harness)
#define NB      16
#define NC      3
#define NPLANE  (NB * NC)            // 48 (b,c) planes
#define HW      (1024 * 1024)        // elements per plane
#define HW4     (HW / 4)             // float4s per plane = 262144

#define BLOCKS_PER_PLANE 128
#define THREADS          256
#define STRIDE4          (BLOCKS_PER_PLANE * THREADS)   // 32768 float4s
#define ITERS            (HW4 / STRIDE4)                // 8, compile-time

typedef float v2f __attribute__((ext_vector_type(2)));
typedef float v4f __attribute__((ext_vector_type(4)));

// ---------------------------------------------------------------------------
// Streaming affine: out = w*img + bias per plane.  (Emitted FIRST so the
// disasm snippet shows the hot loop.)
// 128-bit non-temporal loads/stores, v_pk_fma_f32 (CDNA5 VOP3P) math,
// fully unrolled 8-deep for memory-level parallelism.
// blockIdx.y = plane, blockIdx.x = chunk.
// ---------------------------------------------------------------------------
__global__ void __launch_bounds__(THREADS)
colorcal_stream_kernel(const float* __restrict__ img,
                       const float2* __restrict__ wb,
                       float* __restrict__ out)
{
    const int plane = blockIdx.y;              // 0..47, uniform
    const float2 p = wb[plane];                // uniform 8B load
    v2f wv; wv.x = p.x; wv.y = p.x;            // {w, w}
    v2f bv; bv.x = p.y; bv.y = p.y;            // {bias, bias}

    const size_t planeBase = (size_t)plane * HW;
    const v4f* __restrict__ src = (const v4f*)(img + planeBase);
    v4f* __restrict__ dst       = (v4f*)(out + planeBase);

    int i = blockIdx.x * THREADS + threadIdx.x;   // float4 index within plane

#pragma unroll
    for (int k = 0; k < ITERS; ++k, i += STRIDE4) {
        v4f v = __builtin_nontemporal_load(&src[i]);  // global_load_b128 NT
        v2f lo; lo.x = v.x; lo.y = v.y;
        v2f hi; hi.x = v.z; hi.y = v.w;
        // Packed f32 FMA (VOP3P), in-place: lo = w*lo + b  (2 FMAs/inst)
        asm("v_pk_fma_f32 %0, %1, %0, %2" : "+v"(lo) : "v"(wv), "v"(bv));
        asm("v_pk_fma_f32 %0, %1, %0, %2" : "+v"(hi) : "v"(wv), "v"(bv));
        v4f r; r.x = lo.x; r.y = lo.y; r.z = hi.x; r.w = hi.y;
        __builtin_nontemporal_store(r, &dst[i]);      // global_store_b128 NT
    }
}

// ---------------------------------------------------------------------------
// Setup: gather per-(batch,channel) scale/bias from the lookup tables.
// 48 outputs total -> one tiny block. ws layout: float2 {w, bias} per plane.
// ---------------------------------------------------------------------------
__global__ void colorcal_setup_kernel(
    const int* __restrict__ camindex,
    const int* __restrict__ idindex,
    const int* __restrict__ dstype,
    const float* __restrict__ wcam1, const float* __restrict__ bcam1,
    const float* __restrict__ wident1, const float* __restrict__ bident1,
    const float* __restrict__ wcam2, const float* __restrict__ bcam2,
    const float* __restrict__ wident2, const float* __restrict__ bident2,
    float2* __restrict__ wb)
{
    int t = threadIdx.x;
    if (t >= NPLANE) return;
    int b = t / NC;
    int c = t % NC;
    int cam = camindex[b];
    int id  = idindex[b];
    int ds  = dstype[b];

    float w, bias;
    if (ds == 0) {
        w    = wcam1[cam * 3 + c] + wident1[id * 3 + c];
        bias = bcam1[cam * 3 + c] + bident1[id * 3 + c];
    } else {
        w    = wcam2[cam * 3 + c] + wident2[id * 3 + c];
        bias = bcam2[cam * 3 + c] + bident2[id * 3 + c];
    }
    wb[t] = make_float2(w, bias);
}

// ---------------------------------------------------------------------------
// Launch
// ---------------------------------------------------------------------------
extern "C" void kernel_launch(void* const* d_in, const int* in_sizes, int n_in,
                              void* d_out, int out_size, void* d_ws, size_t ws_size,
                              hipStream_t stream)
{
    const float* image    = (const float*)d_in[0];
    const int*   camindex = (const int*)  d_in[1];
    const int*   idindex  = (const int*)  d_in[2];
    const int*   dstype   = (const int*)  d_in[3];
    const float* wcam1    = (const float*)d_in[4];
    const float* bcam1    = (const float*)d_in[5];
    const float* wident1  = (const float*)d_in[6];
    const float* bident1  = (const float*)d_in[7];
    const float* wcam2    = (const float*)d_in[8];
    const float* bcam2    = (const float*)d_in[9];
    const float* wident2  = (const float*)d_in[10];
    const float* bident2  = (const float*)d_in[11];

    float2* wb = (float2*)d_ws;          // 48 * 8 bytes of scratch
    float*  out = (float*)d_out;

    colorcal_setup_kernel<<<1, 64, 0, stream>>>(
        camindex, idindex, dstype,
        wcam1, bcam1, wident1, bident1,
        wcam2, bcam2, wident2, bident2, wb);

    dim3 grid(BLOCKS_PER_PLANE, NPLANE);
    colorcal_stream_kernel<<<grid, THREADS, 0, stream>>>(image, wb, out);
}